// GeneDynamics_12206297055731
// MI455X (gfx1250) — compile-verified
//
#include <hip/hip_runtime.h>
#include <hip/hip_bf16.h>

typedef __attribute__((ext_vector_type(2))) float v2f;
typedef __attribute__((ext_vector_type(8))) float v8f;

#define N_NODES 16384
#define DIM 16

// ---------------------------------------------------------------------------
// Kernel 1: xhT[d][n] = x[n][d]^2   (squared activation, transposed so the
// WMMA B-operand becomes a contiguous per-lane b64 load along K)
// ---------------------------------------------------------------------------
__global__ __launch_bounds__(256) void xpow_transpose_kernel(
    const float* __restrict__ x, float* __restrict__ xhT) {
  int idx = blockIdx.x * blockDim.x + threadIdx.x;   // 0 .. N_NODES*DIM-1
  int n = idx / DIM;
  int d = idx % DIM;
  float v = x[idx];
  xhT[(size_t)d * N_NODES + n] = v * v;
}

// ---------------------------------------------------------------------------
// Kernel 2: one wave32 per 16-row tile of A.
//   D[16x16] = sum_K  A[m0:m0+16, k:k+4] * xhT^T[k:k+4, 0:16]
// using V_WMMA_F32_16X16X4_F32, then epilogue -x + D/(x^2+1).
//
// A-operand layout (f32 16x4): lane l in [0,16): row=l, K={0,1} in 2 VGPRs;
// lanes [16,32): row=l-16, K={2,3}.  -> per-lane b64 at
//   A + (m0 + lane%16)*16384 + k + 2*(lane/16)
// B-operand mirrors it with column = lane%16 over xhT rows.
// ---------------------------------------------------------------------------
__global__ __launch_bounds__(256) void gene_dyn_wmma_kernel(
    const float* __restrict__ A, const float* __restrict__ x,
    const float* __restrict__ xhT, float* __restrict__ out) {
  const int lane = threadIdx.x & 31;
  const int wave = threadIdx.x >> 5;
  const int tile = blockIdx.x * 8 + wave;   // 16-row tile index (0..1023)
  const int m0   = tile * 16;
  const int lm   = lane & 15;               // A: row-in-tile, B: column (N)
  const int kq   = (lane >> 4) * 2;         // K sub-offset within quad: 0 or 2

  const float* __restrict__ arow = A   + (size_t)(m0 + lm) * N_NODES + kq;
  const float* __restrict__ brow = xhT + (size_t)lm        * N_NODES + kq;

  v8f c = {};

#pragma unroll 1
  for (int k = 0; k < N_NODES; k += 32) {
    // stream-prefetch A ~1KB/row ahead (emits global_prefetch_b8)
    if (k + 256 < N_NODES) {
      __builtin_prefetch(arow + k + 256, 0, 1);
    }

    v2f a0 = *(const v2f*)(arow + k +  0);
    v2f a1 = *(const v2f*)(arow + k +  4);
    v2f a2 = *(const v2f*)(arow + k +  8);
    v2f a3 = *(const v2f*)(arow + k + 12);
    v2f a4 = *(const v2f*)(arow + k + 16);
    v2f a5 = *(const v2f*)(arow + k + 20);
    v2f a6 = *(const v2f*)(arow + k + 24);
    v2f a7 = *(const v2f*)(arow + k + 28);

    v2f b0 = *(const v2f*)(brow + k +  0);
    v2f b1 = *(const v2f*)(brow + k +  4);
    v2f b2 = *(const v2f*)(brow + k +  8);
    v2f b3 = *(const v2f*)(brow + k + 12);
    v2f b4 = *(const v2f*)(brow + k + 16);
    v2f b5 = *(const v2f*)(brow + k + 20);
    v2f b6 = *(const v2f*)(brow + k + 24);
    v2f b7 = *(const v2f*)(brow + k + 28);

    // 8 args: (neg_a, A, neg_b, B, c_mod, C, reuse_a, reuse_b)
    c = __builtin_amdgcn_wmma_f32_16x16x4_f32(false, a0, false, b0, (short)0, c, false, false);
    c = __builtin_amdgcn_wmma_f32_16x16x4_f32(false, a1, false, b1, (short)0, c, false, false);
    c = __builtin_amdgcn_wmma_f32_16x16x4_f32(false, a2, false, b2, (short)0, c, false, false);
    c = __builtin_amdgcn_wmma_f32_16x16x4_f32(false, a3, false, b3, (short)0, c, false, false);
    c = __builtin_amdgcn_wmma_f32_16x16x4_f32(false, a4, false, b4, (short)0, c, false, false);
    c = __builtin_amdgcn_wmma_f32_16x16x4_f32(false, a5, false, b5, (short)0, c, false, false);
    c = __builtin_amdgcn_wmma_f32_16x16x4_f32(false, a6, false, b6, (short)0, c, false, false);
    c = __builtin_amdgcn_wmma_f32_16x16x4_f32(false, a7, false, b7, (short)0, c, false, false);
  }

  // Epilogue: C/D layout -> VGPR v: lanes 0-15 hold M=v, lanes 16-31 M=v+8,
  // N = lane%16.  out[m][n] = -x[m][n] + D[m][n] / (x[m][n]^2 + 1)
  const int col   = lm;
  const int rbase = m0 + (lane >> 4) * 8;
#pragma unroll
  for (int v = 0; v < 8; ++v) {
    const int row = rbase + v;
    const float xv = x[(size_t)row * DIM + col];
    const float xh = xv * xv;
    out[(size_t)row * DIM + col] = -xv + c[v] / (xh + 1.0f);
  }
}

// ---------------------------------------------------------------------------
extern "C" void kernel_launch(void* const* d_in, const int* in_sizes, int n_in,
                              void* d_out, int out_size, void* d_ws, size_t ws_size,
                              hipStream_t stream) {
  const float* A = (const float*)d_in[0];   // [16384, 16384] f32
  const float* x = (const float*)d_in[1];   // [16384, 16] f32
  // d_in[2] = t (unused)
  float* out = (float*)d_out;               // [16384, 16] f32
  float* xhT = (float*)d_ws;                // [16, 16384] f32 = 1 MB scratch

  xpow_transpose_kernel<<<(N_NODES * DIM) / 256, 256, 0, stream>>>(x, xhT);
  gene_dyn_wmma_kernel<<<128, 256, 0, stream>>>(A, x, xhT, out);
}